// SSIM_47888885350997
// MI455X (gfx1250) — compile-verified
//
#include <hip/hip_runtime.h>

// SSIM, fp32, NCHW (8,3,384,1280). Memory-bound 3x3 stencil:
//   out(i,j) = clip(0.5*(1 - n/d), 0, 1) on interior, 0 on the 1-pixel ring.
// CDNA5 path: async global->LDS tile staging (ASYNCcnt) + LDS-separable filter.

#define IMG_H 384
#define IMG_W 1280
#define TILE_H 16
#define TILE_W 64
#define LDS_R (TILE_H + 2)   // 18
#define LDS_C (TILE_W + 2)   // 66

__launch_bounds__(256)
__global__ void ssim_kernel(const float* __restrict__ x,
                            const float* __restrict__ y,
                            float* __restrict__ out)
{
    __shared__ float xs[LDS_R * LDS_C];
    __shared__ float ys[LDS_R * LDS_C];

    const int tid = threadIdx.x;
    const int r0  = blockIdx.y * TILE_H;
    const int c0  = blockIdx.x * TILE_W;
    const long imgbase = (long)blockIdx.z * (IMG_H * IMG_W);

    const float* xi = x + imgbase;
    const float* yi = y + imgbase;

    const unsigned xs_base = (unsigned)(size_t)(&xs[0]);
    const unsigned ys_base = (unsigned)(size_t)(&ys[0]);

    // ---- Stage 1: async DMA the (TILE_H+2)x(TILE_W+2) halo tiles into LDS ----
    for (int k = tid; k < LDS_R * LDS_C; k += 256) {
        const int lr = k / LDS_C;
        const int lc = k - lr * LDS_C;
        int gr = r0 - 1 + lr; gr = gr < 0 ? 0 : (gr > IMG_H - 1 ? IMG_H - 1 : gr);
        int gc = c0 - 1 + lc; gc = gc < 0 ? 0 : (gc > IMG_W - 1 ? IMG_W - 1 : gc);
        const float* gx = xi + (gr * IMG_W + gc);
        const float* gy = yi + (gr * IMG_W + gc);
        const unsigned lx = xs_base + (unsigned)(k * 4);
        const unsigned ly = ys_base + (unsigned)(k * 4);
        // gfx1250 async global -> LDS copy (tracked by ASYNCcnt)
        asm volatile("global_load_async_to_lds_b32 %0, %1, off"
                     :: "v"(lx), "v"(gx) : "memory");
        asm volatile("global_load_async_to_lds_b32 %0, %1, off"
                     :: "v"(ly), "v"(gy) : "memory");
    }
#if __has_builtin(__builtin_amdgcn_s_wait_asynccnt)
    __builtin_amdgcn_s_wait_asynccnt(0);
#else
    asm volatile("s_wait_asynccnt 0" ::: "memory");
#endif
    __syncthreads();

    // ---- Stage 2: separable 3x3 sums + SSIM formula ----
    const int col = tid & (TILE_W - 1);   // 0..63
    const int rg  = tid >> 6;             // 0..3
    const int rb  = rg * 4;               // local output row base

    const float C1   = 1.0e-4f;           // 0.01^2
    const float C2   = 9.0e-4f;           // 0.03^2
    const float inv9 = 1.0f / 9.0f;

    float hx[3], hy[3], hxx[3], hyy[3], hxy[3];

#pragma unroll
    for (int t = 0; t < 6; ++t) {
        const int li = rb + t;            // LDS tile input row
        const float* xr = &xs[li * LDS_C + col];
        const float* yr = &ys[li * LDS_C + col];
        const float a0 = xr[0], a1 = xr[1], a2 = xr[2];
        const float b0 = yr[0], b1 = yr[1], b2 = yr[2];

        if (t >= 3) {  // slide the 3-row window
            hx[0]=hx[1];  hx[1]=hx[2];
            hy[0]=hy[1];  hy[1]=hy[2];
            hxx[0]=hxx[1]; hxx[1]=hxx[2];
            hyy[0]=hyy[1]; hyy[1]=hyy[2];
            hxy[0]=hxy[1]; hxy[1]=hxy[2];
        }
        const int w = (t < 3) ? t : 2;
        hx[w]  = a0 + a1 + a2;
        hy[w]  = b0 + b1 + b2;
        hxx[w] = a0*a0 + a1*a1 + a2*a2;
        hyy[w] = b0*b0 + b1*b1 + b2*b2;
        hxy[w] = a0*b0 + a1*b1 + a2*b2;

        if (t >= 2) {
            const int r  = rb + (t - 2);          // local output row
            const int gi = r0 + r;
            const int gj = c0 + col;
            float v = 0.0f;
            if (gi >= 1 && gi <= IMG_H - 2 && gj >= 1 && gj <= IMG_W - 2) {
                const float sx  = hx[0]  + hx[1]  + hx[2];
                const float sy  = hy[0]  + hy[1]  + hy[2];
                const float sxx = hxx[0] + hxx[1] + hxx[2];
                const float syy = hyy[0] + hyy[1] + hyy[2];
                const float sxy = hxy[0] + hxy[1] + hxy[2];
                const float mux = sx * inv9;
                const float muy = sy * inv9;
                const float mxy = mux * muy;
                const float mxx = mux * mux;
                const float myy = muy * muy;
                const float sigx  = sxx * inv9 - mxx;
                const float sigy  = syy * inv9 - myy;
                const float sigxy = sxy * inv9 - mxy;
                const float n = (2.0f * mxy + C1) * (2.0f * sigxy + C2);
                const float d = (mxx + myy + C1) * (sigx + sigy + C2);
                v = (1.0f - n / d) * 0.5f;
                v = v < 0.0f ? 0.0f : (v > 1.0f ? 1.0f : v);
            }
            out[imgbase + (long)gi * IMG_W + gj] = v;
        }
    }
}

extern "C" void kernel_launch(void* const* d_in, const int* in_sizes, int n_in,
                              void* d_out, int out_size, void* d_ws, size_t ws_size,
                              hipStream_t stream) {
    (void)in_sizes; (void)n_in; (void)out_size; (void)d_ws; (void)ws_size;
    const float* x = (const float*)d_in[0];
    const float* y = (const float*)d_in[1];
    float* out = (float*)d_out;

    dim3 grid(IMG_W / TILE_W,   // 20 column tiles
              IMG_H / TILE_H,   // 24 row tiles
              8 * 3);           // N*C image planes
    dim3 block(256);
    ssim_kernel<<<grid, block, 0, stream>>>(x, y, out);
}